// Warp_1949915152541
// MI455X (gfx1250) — compile-verified
//
#include <hip/hip_runtime.h>
#include <hip/hip_bf16.h>

#define IMG_H 1080
#define IMG_W 1920

typedef float v2f __attribute__((ext_vector_type(2)));
typedef float v8f __attribute__((ext_vector_type(8)));

// ---------------------------------------------------------------------------
// 4x4 inverse (cofactor expansion) -- run by a single lane in the prep kernel
// ---------------------------------------------------------------------------
__device__ inline void inv4x4(const float* m, float* inv) {
    float i0  =  m[5]*m[10]*m[15] - m[5]*m[11]*m[14] - m[9]*m[6]*m[15] + m[9]*m[7]*m[14] + m[13]*m[6]*m[11] - m[13]*m[7]*m[10];
    float i4  = -m[4]*m[10]*m[15] + m[4]*m[11]*m[14] + m[8]*m[6]*m[15] - m[8]*m[7]*m[14] - m[12]*m[6]*m[11] + m[12]*m[7]*m[10];
    float i8  =  m[4]*m[9]*m[15]  - m[4]*m[11]*m[13] - m[8]*m[5]*m[15] + m[8]*m[7]*m[13] + m[12]*m[5]*m[11] - m[12]*m[7]*m[9];
    float i12 = -m[4]*m[9]*m[14]  + m[4]*m[10]*m[13] + m[8]*m[5]*m[14] - m[8]*m[6]*m[13] - m[12]*m[5]*m[10] + m[12]*m[6]*m[9];
    float i1  = -m[1]*m[10]*m[15] + m[1]*m[11]*m[14] + m[9]*m[2]*m[15] - m[9]*m[3]*m[14] - m[13]*m[2]*m[11] + m[13]*m[3]*m[10];
    float i5  =  m[0]*m[10]*m[15] - m[0]*m[11]*m[14] - m[8]*m[2]*m[15] + m[8]*m[3]*m[14] + m[12]*m[2]*m[11] - m[12]*m[3]*m[10];
    float i9  = -m[0]*m[9]*m[15]  + m[0]*m[11]*m[13] + m[8]*m[1]*m[15] - m[8]*m[3]*m[13] - m[12]*m[1]*m[11] + m[12]*m[3]*m[9];
    float i13 =  m[0]*m[9]*m[14]  - m[0]*m[10]*m[13] - m[8]*m[1]*m[14] + m[8]*m[2]*m[13] + m[12]*m[1]*m[10] - m[12]*m[2]*m[9];
    float i2  =  m[1]*m[6]*m[15]  - m[1]*m[7]*m[14]  - m[5]*m[2]*m[15] + m[5]*m[3]*m[14] + m[13]*m[2]*m[7]  - m[13]*m[3]*m[6];
    float i6  = -m[0]*m[6]*m[15]  + m[0]*m[7]*m[14]  + m[4]*m[2]*m[15] - m[4]*m[3]*m[14] - m[12]*m[2]*m[7]  + m[12]*m[3]*m[6];
    float i10 =  m[0]*m[5]*m[15]  - m[0]*m[7]*m[13]  - m[4]*m[1]*m[15] + m[4]*m[3]*m[13] + m[12]*m[1]*m[7]  - m[12]*m[3]*m[5];
    float i14 = -m[0]*m[5]*m[14]  + m[0]*m[6]*m[13]  + m[4]*m[1]*m[14] - m[4]*m[2]*m[13] - m[12]*m[1]*m[6]  + m[12]*m[2]*m[5];
    float i3  = -m[1]*m[6]*m[11]  + m[1]*m[7]*m[10]  + m[5]*m[2]*m[11] - m[5]*m[3]*m[10] - m[9]*m[2]*m[7]   + m[9]*m[3]*m[6];
    float i7  =  m[0]*m[6]*m[11]  - m[0]*m[7]*m[10]  - m[4]*m[2]*m[11] + m[4]*m[3]*m[10] + m[8]*m[2]*m[7]   - m[8]*m[3]*m[6];
    float i11 = -m[0]*m[5]*m[11]  + m[0]*m[7]*m[9]   + m[4]*m[1]*m[11] - m[4]*m[3]*m[9]  - m[8]*m[1]*m[7]   + m[8]*m[3]*m[5];
    float i15 =  m[0]*m[5]*m[10]  - m[0]*m[6]*m[9]   - m[4]*m[1]*m[10] + m[4]*m[2]*m[9]  + m[8]*m[1]*m[6]   - m[8]*m[2]*m[5];
    float det = m[0]*i0 + m[1]*i4 + m[2]*i8 + m[3]*i12;
    float id  = 1.0f / det;
    inv[0]=i0*id;  inv[1]=i1*id;  inv[2]=i2*id;  inv[3]=i3*id;
    inv[4]=i4*id;  inv[5]=i5*id;  inv[6]=i6*id;  inv[7]=i7*id;
    inv[8]=i8*id;  inv[9]=i9*id;  inv[10]=i10*id; inv[11]=i11*id;
    inv[12]=i12*id; inv[13]=i13*id; inv[14]=i14*id; inv[15]=i15*id;
}

// ---------------------------------------------------------------------------
// 4x4 matmul on the WMMA unit: out = Amat * Bmat  (4x4 blocks of a 16x16x4 op)
// Single wave32, EXEC all ones at the WMMA itself. Matrices row-major in LDS.
// A layout (16x4 f32): lanes 0-15 hold K=0 (v0), K=1 (v1); lanes 16-31 K=2/K=3.
// B layout (4x16 f32): v0 = rows K=0 (lanes 0-15) / K=2 (lanes 16-31); v1 = K=1/K=3.
// D layout (16x16 f32): lane n (0-15) holds col n; vgpr r = row r (lanes 0-15).
// ---------------------------------------------------------------------------
__device__ inline void mul4x4_wmma(float* out, const float* Amat, const float* Bmat) {
    const int lane = threadIdx.x & 31;
    const int m  = lane & 15;     // row for A / col for B
    const int kh = lane >> 4;     // K half: 0 -> K{0,1}, 1 -> K{2,3}
    v2f a = {0.0f, 0.0f};
    v2f b = {0.0f, 0.0f};
    if (m < 4) {
        a.x = Amat[m * 4 + 2 * kh + 0];
        a.y = Amat[m * 4 + 2 * kh + 1];
        b.x = Bmat[(2 * kh + 0) * 4 + m];
        b.y = Bmat[(2 * kh + 1) * 4 + m];
    }
    v8f c = {};
    c = __builtin_amdgcn_wmma_f32_16x16x4_f32(
            /*neg_a=*/false, a, /*neg_b=*/false, b,
            /*c_mod=*/(short)0, c, /*reuse_a=*/false, /*reuse_b=*/false);
    if (lane < 4) {               // rows 0..3 live in vgprs 0..3, col = lane
        out[0 * 4 + lane] = c[0];
        out[1 * 4 + lane] = c[1];
        out[2 * 4 + lane] = c[2];
        out[3 * 4 + lane] = c[3];
    }
    __syncthreads();
}

// ---------------------------------------------------------------------------
// Prep kernel (1 wave): P(3x4) = [ K*R*Kinv | K*t ] via K4 * (F1*F2inv) * K4inv
// ---------------------------------------------------------------------------
__global__ void warp_prep_kernel(const float* __restrict__ f1,
                                 const float* __restrict__ f2,
                                 const float* __restrict__ Km,
                                 float* __restrict__ Pout) {
    __shared__ float sF1[16], sF2i[16], sK4[16], sK4i[16], sT[16], sKT[16], sM4[16];
    const int lane = threadIdx.x;

    if (lane == 0) {
        float f2l[16];
        #pragma unroll
        for (int i = 0; i < 16; ++i) { sF1[i] = f1[i]; f2l[i] = f2[i]; }
        float f2inv[16];
        inv4x4(f2l, f2inv);
        #pragma unroll
        for (int i = 0; i < 16; ++i) sF2i[i] = f2inv[i];

        // K embedded as 4x4 (K3 top-left, 1 at [3][3])
        float a = Km[0], bb = Km[1], cc = Km[2];
        float d = Km[3], e  = Km[4], ff = Km[5];
        float g = Km[6], h  = Km[7], i  = Km[8];
        #pragma unroll
        for (int t = 0; t < 16; ++t) { sK4[t] = 0.0f; sK4i[t] = 0.0f; }
        sK4[0]=a;  sK4[1]=bb; sK4[2]=cc;
        sK4[4]=d;  sK4[5]=e;  sK4[6]=ff;
        sK4[8]=g;  sK4[9]=h;  sK4[10]=i;
        sK4[15]=1.0f;
        // analytic 3x3 inverse of K
        float det = a*(e*i - ff*h) - bb*(d*i - ff*g) + cc*(d*h - e*g);
        float id  = 1.0f / det;
        sK4i[0]  =  (e*i  - ff*h) * id;
        sK4i[1]  = -(bb*i - cc*h) * id;
        sK4i[2]  =  (bb*ff - cc*e) * id;
        sK4i[4]  = -(d*i  - ff*g) * id;
        sK4i[5]  =  (a*i  - cc*g) * id;
        sK4i[6]  = -(a*ff - cc*d) * id;
        sK4i[8]  =  (d*h  - e*g) * id;
        sK4i[9]  = -(a*h  - bb*g) * id;
        sK4i[10] =  (a*e  - bb*d) * id;
        sK4i[15] = 1.0f;
    }
    __syncthreads();

    mul4x4_wmma(sT,  sF1, sF2i);   // T  = f1 * inv(f2)
    mul4x4_wmma(sKT, sK4, sT);     // KT = K4 * T
    mul4x4_wmma(sM4, sKT, sK4i);   // M4 = KT * K4inv

    if (lane < 12) Pout[lane] = sM4[lane];   // P = M4[:3, :4], row-major
}

// ---------------------------------------------------------------------------
// Main kernel: one thread per pixel. Projection math + bilinear gather.
// ---------------------------------------------------------------------------
__global__ __launch_bounds__(256) void warp_main_kernel(
        const float* __restrict__ P,
        const float* __restrict__ c1,   // [H, W, 3]
        const float* __restrict__ d1,   // [H, W]
        float* __restrict__ out) {
    const int w = blockIdx.x * blockDim.x + threadIdx.x;
    const int h = blockIdx.y;
    if (w >= IMG_W) return;

    // P is uniform across the wave -> scalar loads
    const float p0 = P[0], p1 = P[1], p2  = P[2],  p3  = P[3];
    const float p4 = P[4], p5 = P[5], p6  = P[6],  p7  = P[7];
    const float p8 = P[8], p9 = P[9], p10 = P[10], p11 = P[11];

    const size_t pix = (size_t)h * IMG_W + w;
    const float d = d1[pix];

    const float xu = (float)w * d;
    const float yv = (float)h * d;
    const float q0 = fmaf(p0, xu, fmaf(p1, yv, fmaf(p2,  d, p3)));
    const float q1 = fmaf(p4, xu, fmaf(p5, yv, fmaf(p6,  d, p7)));
    const float q2 = fmaf(p8, xu, fmaf(p9, yv, fmaf(p10, d, p11)));

    const float invz = 1.0f / q2;
    const float wx = q0 * invz;
    const float wy = q1 * invz;
    const float nx = fmaf(wx, 2.0f / IMG_W, -1.0f);
    const float ny = fmaf(wy, 2.0f / IMG_H, -1.0f);

    // bilinear sample coords (align_corners=False, zeros padding)
    const float sx = (nx + 1.0f) * (0.5f * IMG_W) - 0.5f;
    const float sy = (ny + 1.0f) * (0.5f * IMG_H) - 0.5f;
    const float x0f = floorf(sx), y0f = floorf(sy);
    const float wx1 = sx - x0f, wx0 = 1.0f - wx1;
    const float wy1 = sy - y0f, wy0 = 1.0f - wy1;
    const int x0 = (int)x0f, y0 = (int)y0f;

    float r0 = 0.0f, r1 = 0.0f, r2 = 0.0f;
    auto tap = [&](int yi, int xi, float wgt) {
        const bool valid = (xi >= 0) & (xi < IMG_W) & (yi >= 0) & (yi < IMG_H);
        const int xc = xi < 0 ? 0 : (xi > IMG_W - 1 ? IMG_W - 1 : xi);
        const int yc = yi < 0 ? 0 : (yi > IMG_H - 1 ? IMG_H - 1 : yi);
        const float* tp = c1 + ((size_t)yc * IMG_W + xc) * 3;  // contiguous 12B -> b96
        const float m = valid ? wgt : 0.0f;
        r0 = fmaf(tp[0], m, r0);
        r1 = fmaf(tp[1], m, r1);
        r2 = fmaf(tp[2], m, r2);
    };
    tap(y0,     x0,     wy0 * wx0);
    tap(y0,     x0 + 1, wy0 * wx1);
    tap(y0 + 1, x0,     wy1 * wx0);
    tap(y0 + 1, x0 + 1, wy1 * wx1);

    // outputs: result | normalized_warps | keep_mask (concatenated flat)
    float* out_res = out;
    float* out_nw  = out + (size_t)IMG_H * IMG_W * 3;
    float* out_msk = out + (size_t)IMG_H * IMG_W * 5;

    out_res[pix * 3 + 0] = r0;
    out_res[pix * 3 + 1] = r1;
    out_res[pix * 3 + 2] = r2;
    out_nw[pix * 2 + 0] = nx;
    out_nw[pix * 2 + 1] = ny;
    const bool keep = (nx < 1.0f) & (ny < 1.0f) & (nx > -1.0f) & (ny > -1.0f);
    out_msk[pix] = keep ? 1.0f : 0.0f;
}

extern "C" void kernel_launch(void* const* d_in, const int* in_sizes, int n_in,
                              void* d_out, int out_size, void* d_ws, size_t ws_size,
                              hipStream_t stream) {
    const float* f1 = (const float*)d_in[0];   // f1_pose  [4,4]
    const float* f2 = (const float*)d_in[1];   // f2_pose  [4,4]
    const float* K  = (const float*)d_in[2];   // K        [3,3]
    // d_in[3] is _K (duplicate of K)
    const float* c1 = (const float*)d_in[4];   // [H,W,3]
    const float* d1 = (const float*)d_in[5];   // [H,W]

    float* P = (float*)d_ws;                   // 12 floats of scratch

    warp_prep_kernel<<<1, 32, 0, stream>>>(f1, f2, K, P);

    dim3 block(256, 1, 1);
    dim3 grid((IMG_W + 255) / 256, IMG_H, 1);
    warp_main_kernel<<<grid, block, 0, stream>>>(P, c1, d1, (float*)d_out);
}